// MemoryBank_5385888989248
// MI455X (gfx1250) — compile-verified
//
#include <hip/hip_runtime.h>
#include <stdint.h>

#define NUM_CLASSES 100
#define QUEUE_SIZE  256
#define FEATURE_DIM 1024
#define BATCH       4096
#define NROWS       (NUM_CLASSES * QUEUE_SIZE)   // 25600 output rows of 1024 f32

// ---------------------------------------------------------------------------
// Kernel A: per-embedding inverse L2 norm (f32, eps-clamped like torch
// F.normalize) + init of the inverse scatter map to -1.
// 4096 blocks x 256 threads; each thread handles one float4 of the row.
// ---------------------------------------------------------------------------
__global__ __launch_bounds__(256) void normsInit(const float* __restrict__ emb,
                                                 float* __restrict__ inv,
                                                 int* __restrict__ map) {
    __shared__ float red[256];
    const int b = blockIdx.x;
    const int t = threadIdx.x;

    const float4* row = reinterpret_cast<const float4*>(emb + (size_t)b * FEATURE_DIM);
    float4 v = row[t];
    red[t] = v.x * v.x + v.y * v.y + v.z * v.z + v.w * v.w;
    __syncthreads();
    #pragma unroll
    for (int off = 128; off > 0; off >>= 1) {
        if (t < off) red[t] += red[t + off];
        __syncthreads();
    }
    if (t == 0) {
        float n = sqrtf(red[0]);
        inv[b] = 1.0f / fmaxf(n, 1e-12f);
    }
    // map init: 4096*256 = 1,048,576 threads cover NROWS (25600)
    const int gt = b * 256 + t;
    if (gt < NROWS) map[gt] = -1;
}

// ---------------------------------------------------------------------------
// Kernel B: serial-FIFO rank per batch element, then inverse map scatter.
// rank[b] = #earlier elements with same label (deterministic, matches the
// reference's tril-sum). 4 blocks x 1024 threads, labels cached in LDS.
// ---------------------------------------------------------------------------
__global__ __launch_bounds__(1024) void rankScatter(const int* __restrict__ labels,
                                                    const int* __restrict__ qptr,
                                                    int* __restrict__ map) {
    __shared__ int lab[BATCH];   // 16 KB
    const int t = threadIdx.x;
    #pragma unroll
    for (int i = t; i < BATCH; i += 1024) lab[i] = labels[i];
    __syncthreads();

    const int b  = blockIdx.x * 1024 + t;   // grid = 4 blocks -> b in [0,4096)
    const int my = lab[b];
    int rank = 0;
    for (int j = 0; j < b; ++j) rank += (lab[j] == my) ? 1 : 0;

    const int slot = (qptr[my] + rank) & (QUEUE_SIZE - 1);   // QUEUE_SIZE is pow2
    // Distinct (class,slot) per element for this input -> conflict-free.
    map[my * QUEUE_SIZE + slot] = b;
}

// ---------------------------------------------------------------------------
// Kernel C: one wave32 per output row (25600 waves = 3200 blocks x 8 waves).
//  - replaced row : direct float4 loads of embedding, scale by inv norm, store
//  - copied row   : CDNA5 async DMA  global -> LDS -> global
//                   (global_load_async_to_lds_b128 / global_store_async_from_lds_b128,
//                    tracked by ASYNCcnt). Branch is wave-uniform, EXEC all-1s.
// 4 KB LDS slice per wave (32 KB / block).
// ---------------------------------------------------------------------------
__global__ __launch_bounds__(256) void writeRows(const float* __restrict__ emb,
                                                 const float* __restrict__ queue,
                                                 const int*   __restrict__ map,
                                                 const float* __restrict__ inv,
                                                 float* __restrict__ out) {
    __shared__ float buf[8][FEATURE_DIM];   // 8 waves * 4KB
    const int t    = threadIdx.x;
    const int wave = t >> 5;
    const int lane = t & 31;
    const int row  = blockIdx.x * 8 + wave;           // < 25600 by construction

    const int b = map[row];                           // wave-uniform
    float* orow = out + (size_t)row * FEATURE_DIM;

    if (b >= 0) {
        // Row replaced by normalized embedding b.
        const float s = inv[b];
        const float4* e = reinterpret_cast<const float4*>(emb + (size_t)b * FEATURE_DIM);
        float4* o = reinterpret_cast<float4*>(orow);
        #pragma unroll
        for (int i = 0; i < 8; ++i) {
            float4 v = e[lane + 32 * i];
            v.x *= s; v.y *= s; v.z *= s; v.w *= s;
            o[lane + 32 * i] = v;
        }
    } else {
        // Bulk copy path: async DMA through this wave's LDS slice.
        const float* qrow = queue + (size_t)row * FEATURE_DIM;
        const unsigned lds_base = (unsigned)(size_t)(&buf[wave][0]);

        #pragma unroll
        for (int i = 0; i < 8; ++i) {
            const unsigned loff = lds_base + (unsigned)((lane + 32 * i) * 16);
            const unsigned long long ga =
                (unsigned long long)(size_t)qrow + (unsigned long long)((lane + 32 * i) * 16);
            asm volatile("global_load_async_to_lds_b128 %0, %1, off"
                         :: "v"(loff), "v"(ga) : "memory");
        }
        asm volatile("s_wait_asynccnt 0" ::: "memory");
        #pragma unroll
        for (int i = 0; i < 8; ++i) {
            const unsigned loff = lds_base + (unsigned)((lane + 32 * i) * 16);
            const unsigned long long ga =
                (unsigned long long)(size_t)orow + (unsigned long long)((lane + 32 * i) * 16);
            asm volatile("global_store_async_from_lds_b128 %0, %1, off"
                         :: "v"(ga), "v"(loff) : "memory");
        }
        asm volatile("s_wait_asynccnt 0" ::: "memory");
    }
}

// ---------------------------------------------------------------------------
// Harness entry. Inputs (setup_inputs order):
//   d_in[0] embeddings f32 (4096,1024)
//   d_in[1] labels     int (4096,)
//   d_in[2] queue      f32 (100,256,1024)
//   d_in[3] queue_ptr  int (100,)
// Output: new_queue f32 (100,256,1024)
// Scratch layout: map[NROWS] ints, then inv[BATCH] floats (~118 KB).
// ---------------------------------------------------------------------------
extern "C" void kernel_launch(void* const* d_in, const int* in_sizes, int n_in,
                              void* d_out, int out_size, void* d_ws, size_t ws_size,
                              hipStream_t stream) {
    const float* emb    = (const float*)d_in[0];
    const int*   labels = (const int*)d_in[1];
    const float* queue  = (const float*)d_in[2];
    const int*   qptr   = (const int*)d_in[3];
    float* out = (float*)d_out;

    int*   map = (int*)d_ws;
    float* inv = (float*)((char*)d_ws + (size_t)NROWS * sizeof(int));

    hipLaunchKernelGGL(normsInit,   dim3(BATCH),     dim3(256),  0, stream, emb, inv, map);
    hipLaunchKernelGGL(rankScatter, dim3(4),         dim3(1024), 0, stream, labels, qptr, map);
    hipLaunchKernelGGL(writeRows,   dim3(NROWS / 8), dim3(256),  0, stream, emb, queue, map, inv, out);
}